// GNN_75814762709759
// MI455X (gfx1250) — compile-verified
//
#include <hip/hip_runtime.h>
#include <hip/hip_bf16.h>

typedef __attribute__((ext_vector_type(2))) float v2f;
typedef __attribute__((ext_vector_type(8))) float v8f;

#define N_NODES 50000
#define N_EDGES 800000
#define D_IN    128
#define D_HID   16
#define N_CLS   3

// ---------------- scratch layout (floats) ----------------
// [0, 1,000,000)      : sum1(800000) | deg(50000) | sum2(150000)  <- zeroed
// [1,000,000 ...)     : y1(800000), z1(800000), h(800000), y2(150000), z2(150000)
#define OFF_SUM1 0
#define OFF_DEG  800000
#define OFF_SUM2 850000
#define OFF_Y1   1000000
#define OFF_Z1   1800000
#define OFF_H    2600000
#define OFF_Y2   3400000
#define OFF_Z2   3550000
#define ZERO_N   1000000

__global__ void gnn_zero(float* __restrict__ p, int n) {
  int i = blockIdx.x * blockDim.x + threadIdx.x;
  if (i < n) p[i] = 0.0f;
}

// y1 = x @ W1_l^T, z1 = x @ W1_r^T  via V_WMMA_F32_16X16X4_F32.
// One wave32 per 16-row tile (50000 = 3125 * 16, no tail).
__global__ void gnn_proj1_wmma(const float* __restrict__ x,
                               const float* __restrict__ Wl,
                               const float* __restrict__ Wr,
                               float* __restrict__ y1,
                               float* __restrict__ z1) {
  int wave = (blockIdx.x * blockDim.x + threadIdx.x) >> 5;
  int lane = threadIdx.x & 31;
  int r0 = wave * 16;
  if (r0 >= N_NODES) return;

  int m    = lane & 15;        // row (A) / col (B,D) within tile
  int koff = (lane >> 4) * 2;  // lanes 16-31 carry K+2, K+3

  const float* __restrict__ xrow = x + (size_t)(r0 + m) * D_IN;
  const float* __restrict__ wlr  = Wl + (size_t)m * D_IN;  // B[k][n] = W[n*D_IN + k]
  const float* __restrict__ wrr  = Wr + (size_t)m * D_IN;

  v8f accL = {};
  v8f accR = {};
  for (int k0 = 0; k0 < D_IN; k0 += 4) {
    v2f a, bl, br;
    a.x  = xrow[k0 + koff];
    a.y  = xrow[k0 + koff + 1];
    bl.x = wlr[k0 + koff];
    bl.y = wlr[k0 + koff + 1];
    br.x = wrr[k0 + koff];
    br.y = wrr[k0 + koff + 1];
    accL = __builtin_amdgcn_wmma_f32_16x16x4_f32(false, a, false, bl,
                                                 (short)0, accL, false, false);
    accR = __builtin_amdgcn_wmma_f32_16x16x4_f32(false, a, false, br,
                                                 (short)0, accR, false, false);
  }
  // D layout: VGPR v -> row r0 + v (+8 for lanes 16-31), col = lane&15
  int rbase = r0 + (lane >> 4) * 8;
#pragma unroll
  for (int v = 0; v < 8; ++v) {
    size_t idx = (size_t)(rbase + v) * D_HID + m;
    y1[idx] = accL[v];
    z1[idx] = accR[v];
  }
}

__global__ void gnn_degree(const int* __restrict__ ei, float* __restrict__ deg) {
  int e = blockIdx.x * blockDim.x + threadIdx.x;
  if (e >= N_EDGES) return;
  int dst = ei[N_EDGES + e];
  atomicAdd(&deg[dst], 1.0f);
}

// scatter y1[src] into sum1[dst]; 4 threads per edge, 4 floats each
__global__ void gnn_scatter16(const int* __restrict__ ei,
                              const float* __restrict__ y1,
                              float* __restrict__ sum1) {
  int tid = blockIdx.x * blockDim.x + threadIdx.x;
  if (tid >= N_EDGES * 4) return;
  int e  = tid >> 2;
  int c0 = (tid & 3) * 4;
  int src = ei[e];
  int dst = ei[N_EDGES + e];
  const float* ys = y1 + (size_t)src * D_HID + c0;
  float* ss = sum1 + (size_t)dst * D_HID + c0;
#pragma unroll
  for (int c = 0; c < 4; ++c) atomicAdd(&ss[c], ys[c]);
}

__global__ void gnn_combine1(const float* __restrict__ sum1,
                             const float* __restrict__ deg,
                             const float* __restrict__ z1,
                             const float* __restrict__ b1,
                             float* __restrict__ h) {
  int tid = blockIdx.x * blockDim.x + threadIdx.x;
  if (tid >= N_NODES * D_HID) return;
  int n = tid >> 4;
  int c = tid & 15;
  float rinv = 1.0f / fmaxf(deg[n], 1.0f);
  float v = sum1[tid] * rinv + b1[c] + z1[tid];
  h[tid] = fmaxf(v, 0.0f);
}

// y2 = h @ W2_l^T, z2 = h @ W2_r^T  (16 -> 3, scalar per node)
__global__ void gnn_proj2(const float* __restrict__ h,
                          const float* __restrict__ W2l,
                          const float* __restrict__ W2r,
                          float* __restrict__ y2,
                          float* __restrict__ z2) {
  int n = blockIdx.x * blockDim.x + threadIdx.x;
  if (n >= N_NODES) return;
  float hv[D_HID];
#pragma unroll
  for (int k = 0; k < D_HID; ++k) hv[k] = h[(size_t)n * D_HID + k];
#pragma unroll
  for (int c = 0; c < N_CLS; ++c) {
    float al = 0.0f, ar = 0.0f;
#pragma unroll
    for (int k = 0; k < D_HID; ++k) {
      al = fmaf(hv[k], W2l[c * D_HID + k], al);
      ar = fmaf(hv[k], W2r[c * D_HID + k], ar);
    }
    y2[(size_t)n * N_CLS + c] = al;
    z2[(size_t)n * N_CLS + c] = ar;
  }
}

__global__ void gnn_scatter3(const int* __restrict__ ei,
                             const float* __restrict__ y2,
                             float* __restrict__ sum2) {
  int e = blockIdx.x * blockDim.x + threadIdx.x;
  if (e >= N_EDGES) return;
  int src = ei[e];
  int dst = ei[N_EDGES + e];
#pragma unroll
  for (int c = 0; c < N_CLS; ++c)
    atomicAdd(&sum2[(size_t)dst * N_CLS + c], y2[(size_t)src * N_CLS + c]);
}

__global__ void gnn_final(const float* __restrict__ sum2,
                          const float* __restrict__ deg,
                          const float* __restrict__ z2,
                          const float* __restrict__ b2,
                          float* __restrict__ out) {
  int n = blockIdx.x * blockDim.x + threadIdx.x;
  if (n >= N_NODES) return;
  float rinv = 1.0f / fmaxf(deg[n], 1.0f);
  float l[N_CLS];
#pragma unroll
  for (int c = 0; c < N_CLS; ++c) {
    float v = sum2[(size_t)n * N_CLS + c] * rinv + b2[c] + z2[(size_t)n * N_CLS + c];
    l[c] = fmaxf(v, 0.0f);
  }
  float mx = fmaxf(l[0], fmaxf(l[1], l[2]));
  float s = __expf(l[0] - mx) + __expf(l[1] - mx) + __expf(l[2] - mx);
  float lse = __logf(s);
#pragma unroll
  for (int c = 0; c < N_CLS; ++c)
    out[(size_t)n * N_CLS + c] = l[c] - mx - lse;
}

extern "C" void kernel_launch(void* const* d_in, const int* in_sizes, int n_in,
                              void* d_out, int out_size, void* d_ws, size_t ws_size,
                              hipStream_t stream) {
  const float* x   = (const float*)d_in[0];
  const int*   ei  = (const int*)d_in[1];   // [2, N_EDGES]
  const float* W1l = (const float*)d_in[2];
  const float* b1  = (const float*)d_in[3];
  const float* W1r = (const float*)d_in[4];
  const float* W2l = (const float*)d_in[5];
  const float* b2  = (const float*)d_in[6];
  const float* W2r = (const float*)d_in[7];
  float* out = (float*)d_out;
  float* ws  = (float*)d_ws;

  float* sum1 = ws + OFF_SUM1;
  float* deg  = ws + OFF_DEG;
  float* sum2 = ws + OFF_SUM2;
  float* y1   = ws + OFF_Y1;
  float* z1   = ws + OFF_Z1;
  float* h    = ws + OFF_H;
  float* y2   = ws + OFF_Y2;
  float* z2   = ws + OFF_Z2;

  const int T = 256;
  // 1) zero accumulators (sum1 | deg | sum2 are contiguous)
  gnn_zero<<<(ZERO_N + T - 1) / T, T, 0, stream>>>(ws, ZERO_N);
  // 2) layer-1 projections via f32 WMMA (3125 waves, 8 waves/block)
  gnn_proj1_wmma<<<(3125 + 7) / 8, T, 0, stream>>>(x, W1l, W1r, y1, z1);
  // 3) in-degree counts
  gnn_degree<<<(N_EDGES + T - 1) / T, T, 0, stream>>>(ei, deg);
  // 4) edge scatter of projected features (16-wide)
  gnn_scatter16<<<(N_EDGES * 4 + T - 1) / T, T, 0, stream>>>(ei, y1, sum1);
  // 5) mean + bias + root + ReLU -> h
  gnn_combine1<<<(N_NODES * D_HID + T - 1) / T, T, 0, stream>>>(sum1, deg, z1, b1, h);
  // 6) layer-2 projections (16 -> 3)
  gnn_proj2<<<(N_NODES + T - 1) / T, T, 0, stream>>>(h, W2l, W2r, y2, z2);
  // 7) edge scatter (3-wide)
  gnn_scatter3<<<(N_EDGES + T - 1) / T, T, 0, stream>>>(ei, y2, sum2);
  // 8) mean + bias + root + ReLU + log_softmax
  gnn_final<<<(N_NODES + T - 1) / T, T, 0, stream>>>(sum2, deg, z2, b2, out);
}